// NeuralTuringMachine_47777216201230
// MI455X (gfx1250) — compile-verified
//
#include <hip/hip_runtime.h>
#include <hip/hip_bf16.h>
#include <math.h>

#define N_SLOTS 65536
#define MDIM    512
#define CDIM    512
#define INDIM   256
#define OUTDIM  256
#define EPSF    1e-8f

typedef __attribute__((ext_vector_type(2))) float v2f;
typedef __attribute__((ext_vector_type(8))) float v8f;

#ifndef __has_builtin
#define __has_builtin(x) 0
#endif
#if __has_builtin(__builtin_amdgcn_wmma_f32_16x16x4_f32)
#define NTM_USE_WMMA 1
#else
#define NTM_USE_WMMA 0
#endif

// ---- workspace layout (float offsets) ----
enum : int {
  OFF_HNEW = 0,            // 512  h_new
  OFF_KEY  = 512,          // 512  key + EPS
  OFF_SCAL = 1024,         // [0]=beta [1]=g [2..4]=shift [5]=gamma [6]=nb [7]=sum(key_w)
  OFF_GMAX = 1040,
  OFF_GSUM = 1041,
  OFF_PSUM = 1042,
  OFF_READ = 1056,         // 512  accumulated read vector
  OFF_BMAX = 1568,         // 512  per-block maxes (K2)
  OFF_BSUM = 2080,         // 256  per-block sums (K4)
  OFF_BSUM2= 2336,         // 256  per-block sums (K6)
  OFF_BCOS = 2592,         // 65536: beta*cos, later overwritten in-place by exp()
  OFF_POW  = 2592 + N_SLOTS // 65536: sharpened weights
};

__device__ __forceinline__ float sigm(float x) { return 1.0f / (1.0f + expf(-x)); }
__device__ __forceinline__ float softplusf(float x) { return x > 20.0f ? x : log1pf(expf(x)); }

// ---------------- K1: LSTM cell + read-head projection + scalars ----------------
__global__ __launch_bounds__(256)
void ntm_k1(const float* __restrict__ x, const float* __restrict__ prev_read,
            const float* __restrict__ h, const float* __restrict__ c,
            const float* __restrict__ W_ih, const float* __restrict__ b_ih,
            const float* __restrict__ W_hh, const float* __restrict__ b_hh,
            const float* __restrict__ W_read, const float* __restrict__ b_read,
            float* __restrict__ ws)
{
  __shared__ float s_inp[INDIM + MDIM];   // 768
  __shared__ float s_hin[CDIM];
  __shared__ float s_gates[4 * CDIM];
  __shared__ float s_hn[CDIM];
  __shared__ float s_key[MDIM];
  __shared__ float s_rext[6];
  const int tid = threadIdx.x;

  for (int k = tid; k < INDIM + MDIM; k += 256)
    s_inp[k] = (k < INDIM) ? x[k] : prev_read[k - INDIM];
  for (int k = tid; k < CDIM; k += 256) s_hin[k] = h[k];
  __syncthreads();

  for (int j = tid; j < 4 * CDIM; j += 256) {
    float acc = b_ih[j] + b_hh[j];
    const float* wi = W_ih + (size_t)j * (INDIM + MDIM);
    for (int k = 0; k < INDIM + MDIM; ++k) acc += s_inp[k] * wi[k];
    const float* wh = W_hh + (size_t)j * CDIM;
    for (int k = 0; k < CDIM; ++k) acc += s_hin[k] * wh[k];
    s_gates[j] = acc;
  }
  __syncthreads();

  for (int m = tid; m < CDIM; m += 256) {
    float ig = s_gates[m], fg = s_gates[CDIM + m];
    float gg = s_gates[2 * CDIM + m], og = s_gates[3 * CDIM + m];
    float cn = sigm(fg) * c[m] + sigm(ig) * tanhf(gg);
    float hn = sigm(og) * tanhf(cn);
    s_hn[m] = hn;
    ws[OFF_HNEW + m] = hn;
    ws[OFF_READ + m] = 0.0f;   // re-zero the read accumulator every call
  }
  __syncthreads();

  for (int j = tid; j < MDIM + 6; j += 256) {
    float acc = b_read[j];
    const float* wr = W_read + (size_t)j * CDIM;
    for (int k = 0; k < CDIM; ++k) acc += s_hn[k] * wr[k];
    if (j < MDIM) { float kw = acc + EPSF; s_key[j] = kw; ws[OFF_KEY + j] = kw; }
    else          s_rext[j - MDIM] = acc;
  }
  __syncthreads();

  if (tid == 0) {
    float beta = softplusf(s_rext[0]);
    float g    = sigm(s_rext[1]);
    float m3   = fmaxf(fmaxf(s_rext[2], s_rext[3]), s_rext[4]);
    float e0 = expf(s_rext[2] - m3), e1 = expf(s_rext[3] - m3), e2 = expf(s_rext[4] - m3);
    float es = e0 + e1 + e2;
    float gamma = 1.0f + softplusf(s_rext[5]);
    float ss = 0.f, sq = 0.f;
    for (int m = 0; m < MDIM; ++m) { float kv = s_key[m]; ss += kv; sq += kv * kv; }
    float nb = fmaxf(sqrtf(sq), EPSF);
    ws[OFF_SCAL + 0] = beta;  ws[OFF_SCAL + 1] = g;
    ws[OFF_SCAL + 2] = e0 / es; ws[OFF_SCAL + 3] = e1 / es; ws[OFF_SCAL + 4] = e2 / es;
    ws[OFF_SCAL + 5] = gamma; ws[OFF_SCAL + 6] = nb; ws[OFF_SCAL + 7] = ss;
  }
}

// ---------------- K2: cosine similarity vs all memory rows (WMMA GEMV) ----------------
// One wave handles 16 rows; B tile = memory rows (documented 4x16 f32 B layout:
// lane n (0..15) holds B[0,n]/B[1,n], lane n+16 holds B[2,n]/B[3,n]); A has the key
// chunk in matrix row 0 only (lanes 0 and 16), zeros elsewhere -> D[0,n] = dot(key,row n).
__global__ __launch_bounds__(256)
void ntm_k2_cos(const float* __restrict__ memory, float* __restrict__ ws)
{
  __shared__ float s_key[MDIM];
  __shared__ float s_red[256];
  const int tid = threadIdx.x;
  for (int k = tid; k < MDIM; k += 256) s_key[k] = ws[OFF_KEY + k];
  __syncthreads();

  const float beta = ws[OFF_SCAL + 0];
  const float nb   = ws[OFF_SCAL + 6];
  const float skw  = ws[OFF_SCAL + 7];

  const int wave = tid >> 5, lane = tid & 31;
  const int row  = blockIdx.x * 128 + wave * 16 + (lane & 15);
  const int ksel = (lane >> 4) * 2;                       // 0 for lanes 0-15, 2 for 16-31
  const float amask = ((lane & 15) == 0) ? 1.0f : 0.0f;   // key occupies A row 0 only
  const float* rowp = memory + (size_t)row * MDIM;

  float s1 = 0.f, s2 = 0.f;
#if NTM_USE_WMMA
  v8f acc = {0.f, 0.f, 0.f, 0.f, 0.f, 0.f, 0.f, 0.f};
#else
  float dotp = 0.f;
#endif
  for (int k0 = 0; k0 < MDIM; k0 += 4) {
    if ((k0 & 31) == 0) __builtin_prefetch(rowp + k0 + 128, 0, 0);
    float2 v = *(const float2*)(rowp + k0 + ksel);
    s1 += v.x + v.y;
    s2 += v.x * v.x + v.y * v.y;
#if NTM_USE_WMMA
    v2f a = {s_key[k0 + ksel] * amask, s_key[k0 + ksel + 1] * amask};
    v2f b = {v.x, v.y};
    acc = __builtin_amdgcn_wmma_f32_16x16x4_f32(false, a, false, b, (short)0, acc,
                                                false, false);
#else
    dotp += s_key[k0 + ksel] * v.x + s_key[k0 + ksel + 1] * v.y;
#endif
  }
  s1 += __shfl_xor(s1, 16, 32);
  s2 += __shfl_xor(s2, 16, 32);
#if NTM_USE_WMMA
  float dotv = acc[0];                     // D[0, n] on lanes 0..15
#else
  float dotv = dotp + __shfl_xor(dotp, 16, 32);
#endif

  float bcos = -INFINITY;
  if (lane < 16) {
    float dot = dotv + EPSF * skw;  // exact correction for a = mem + EPS
    float na  = fmaxf(sqrtf(s2 + 2.0f * EPSF * s1 + (float)MDIM * EPSF * EPSF), EPSF);
    bcos = beta * (dot / (na * nb));
    ws[OFF_BCOS + row] = bcos;
  }
  s_red[tid] = bcos;
  __syncthreads();
  for (int s = 128; s > 0; s >>= 1) {
    if (tid < s) s_red[tid] = fmaxf(s_red[tid], s_red[tid + s]);
    __syncthreads();
  }
  if (tid == 0) ws[OFF_BMAX + blockIdx.x] = s_red[0];
}

// ---------------- K3: reduce 512 block maxes ----------------
__global__ __launch_bounds__(256)
void ntm_k3_max(float* __restrict__ ws)
{
  __shared__ float r[256];
  const int tid = threadIdx.x;
  r[tid] = fmaxf(ws[OFF_BMAX + tid], ws[OFF_BMAX + 256 + tid]);
  __syncthreads();
  for (int s = 128; s > 0; s >>= 1) {
    if (tid < s) r[tid] = fmaxf(r[tid], r[tid + s]);
    __syncthreads();
  }
  if (tid == 0) ws[OFF_GMAX] = r[0];
}

// ---------------- K4: exp(bcos - gmax) in place + partial sums ----------------
__global__ __launch_bounds__(256)
void ntm_k4_exp(float* __restrict__ ws)
{
  __shared__ float r[256];
  const int tid = threadIdx.x;
  const int n = blockIdx.x * 256 + tid;
  const float e = expf(ws[OFF_BCOS + n] - ws[OFF_GMAX]);
  ws[OFF_BCOS + n] = e;
  r[tid] = e;
  __syncthreads();
  for (int s = 128; s > 0; s >>= 1) {
    if (tid < s) r[tid] += r[tid + s];
    __syncthreads();
  }
  if (tid == 0) ws[OFF_BSUM + blockIdx.x] = r[0];
}

// ---------------- generic 256-element sum reduce ----------------
__global__ __launch_bounds__(256)
void ntm_reduce_sum(float* __restrict__ ws, int srcOff, int dstOff)
{
  __shared__ float r[256];
  const int tid = threadIdx.x;
  r[tid] = ws[srcOff + tid];
  __syncthreads();
  for (int s = 128; s > 0; s >>= 1) {
    if (tid < s) r[tid] += r[tid + s];
    __syncthreads();
  }
  if (tid == 0) ws[dstOff] = r[0];
}

// ---------------- K6: interpolate, circular 3-tap shift, sharpen ----------------
__global__ __launch_bounds__(256)
void ntm_k6_shiftpow(const float* __restrict__ read_state, float* __restrict__ ws)
{
  __shared__ float r[256];
  const int tid = threadIdx.x;
  const int n = blockIdx.x * 256 + tid;
  const float inv = 1.0f / ws[OFF_GSUM];
  const float g = ws[OFF_SCAL + 1];
  const float c0 = ws[OFF_SCAL + 2], c1 = ws[OFF_SCAL + 3], c2 = ws[OFF_SCAL + 4];
  const float gamma = ws[OFF_SCAL + 5];
  const int nm1 = (n + N_SLOTS - 1) & (N_SLOTS - 1);
  const int np1 = (n + 1) & (N_SLOTS - 1);
  const float wim1 = g * read_state[nm1] + (1.0f - g) * (ws[OFF_BCOS + nm1] * inv);
  const float wi0  = g * read_state[n]   + (1.0f - g) * (ws[OFF_BCOS + n]   * inv);
  const float wip1 = g * read_state[np1] + (1.0f - g) * (ws[OFF_BCOS + np1] * inv);
  const float wsft = wim1 * c0 + wi0 * c1 + wip1 * c2;
  const float p = powf(wsft, gamma);
  ws[OFF_POW + n] = p;
  r[tid] = p;
  __syncthreads();
  for (int s = 128; s > 0; s >>= 1) {
    if (tid < s) r[tid] += r[tid + s];
    __syncthreads();
  }
  if (tid == 0) ws[OFF_BSUM2 + blockIdx.x] = r[0];
}

// ---------------- K8: read = sum_n w[n] * memory[n, :] (coalesced) ----------------
__global__ __launch_bounds__(256)
void ntm_k8_read(const float* __restrict__ memory, float* __restrict__ ws)
{
  const int tid = threadIdx.x;
  const float invp = 1.0f / (ws[OFF_PSUM] + EPSF);
  const int base = blockIdx.x * 128;
  float a0 = 0.f, a1 = 0.f;
  for (int r = 0; r < 128; ++r) {
    const int n = base + r;
    const float w = ws[OFF_POW + n] * invp;
    const float* rp = memory + (size_t)n * MDIM;
    a0 += w * rp[tid];
    a1 += w * rp[256 + tid];
  }
  atomicAdd(&ws[OFF_READ + tid], a0);
  atomicAdd(&ws[OFF_READ + 256 + tid], a1);
}

// ---------------- K9: out = sigmoid([h_new, read] @ W_out^T + b_out) ----------------
__global__ __launch_bounds__(256)
void ntm_k9_out(const float* __restrict__ W_out, const float* __restrict__ b_out,
                const float* __restrict__ ws, float* __restrict__ out)
{
  __shared__ float s_v[CDIM + MDIM];
  const int tid = threadIdx.x;
  for (int k = tid; k < CDIM; k += 256) s_v[k] = ws[OFF_HNEW + k];
  for (int k = tid; k < MDIM; k += 256) s_v[CDIM + k] = ws[OFF_READ + k];
  __syncthreads();
  float acc = b_out[tid];
  const float* w = W_out + (size_t)tid * (CDIM + MDIM);
  for (int k = 0; k < CDIM + MDIM; ++k) acc += s_v[k] * w[k];
  out[tid] = sigm(acc);
}

extern "C" void kernel_launch(void* const* d_in, const int* in_sizes, int n_in,
                              void* d_out, int out_size, void* d_ws, size_t ws_size,
                              hipStream_t stream)
{
  (void)in_sizes; (void)n_in; (void)out_size; (void)ws_size;
  const float* x          = (const float*)d_in[0];
  const float* memory     = (const float*)d_in[1];
  const float* prev_read  = (const float*)d_in[2];
  const float* h          = (const float*)d_in[3];
  const float* c          = (const float*)d_in[4];
  const float* read_state = (const float*)d_in[5];
  // d_in[6] write_state, d_in[13] W_write, d_in[14] b_write: dead code in reference
  const float* W_ih   = (const float*)d_in[7];
  const float* b_ih   = (const float*)d_in[8];
  const float* W_hh   = (const float*)d_in[9];
  const float* b_hh   = (const float*)d_in[10];
  const float* W_read = (const float*)d_in[11];
  const float* b_read = (const float*)d_in[12];
  const float* W_out  = (const float*)d_in[15];
  const float* b_out  = (const float*)d_in[16];
  float* ws  = (float*)d_ws;
  float* out = (float*)d_out;

  ntm_k1<<<1, 256, 0, stream>>>(x, prev_read, h, c, W_ih, b_ih, W_hh, b_hh,
                                W_read, b_read, ws);
  ntm_k2_cos<<<N_SLOTS / 128, 256, 0, stream>>>(memory, ws);
  ntm_k3_max<<<1, 256, 0, stream>>>(ws);
  ntm_k4_exp<<<N_SLOTS / 256, 256, 0, stream>>>(ws);
  ntm_reduce_sum<<<1, 256, 0, stream>>>(ws, OFF_BSUM, OFF_GSUM);
  ntm_k6_shiftpow<<<N_SLOTS / 256, 256, 0, stream>>>(read_state, ws);
  ntm_reduce_sum<<<1, 256, 0, stream>>>(ws, OFF_BSUM2, OFF_PSUM);
  ntm_k8_read<<<N_SLOTS / 128, 256, 0, stream>>>(memory, ws);
  ntm_k9_out<<<1, 256, 0, stream>>>(W_out, b_out, ws, out);
}